// EmbeddingClassifier_45226005627622
// MI455X (gfx1250) — compile-verified
//
#include <hip/hip_runtime.h>

// CDNA5 / gfx1250: wave32, WMMA (not MFMA).
// GEMM logits[32768,64] = A[32768,1024] x W[1024,64] + b, mask -> -inf, argmax.
// Exact fp32 math via V_WMMA_F32_16X16X4_F32 (16x16 f32 tile, K-step 4).
//
// W is pre-packed into K-pair-major form (Wp[j][n] = {W[2j][n], W[2j+1][n]})
// so every B fragment is one coalesced global_load_b64. Each wave computes a
// 32(M) x 64(N) strip: 8 accumulators, 6 b64 loads per 8 WMMAs.

typedef __attribute__((ext_vector_type(2))) float v2f;
typedef __attribute__((ext_vector_type(8))) float v8f;

#define NEG_INF (-__builtin_inff())

constexpr int M_TOTAL = 32 * 1024;   // B*S rows
constexpr int K_DIM   = 1024;        // D
constexpr int N_DIM   = 64;          // C
constexpr int WAVES_PER_BLOCK = 8;   // 256 threads, wave32
constexpr int ROWS_PER_WAVE   = 32;  // two 16-row M-tiles
constexpr int ROWS_PER_BLOCK  = WAVES_PER_BLOCK * ROWS_PER_WAVE; // 256

// ---------------------------------------------------------------------------
// Pack W[1024,64] -> Wp[512][64][2]: adjacent K-pairs per column, so a B
// fragment (VGPR0=K, VGPR1=K+1 at column n) is one aligned b64 load.
// ---------------------------------------------------------------------------
__global__ __launch_bounds__(256) void pack_W(const float* __restrict__ W,
                                              float* __restrict__ Wp) {
  const int idx = blockIdx.x * blockDim.x + threadIdx.x; // 0..32767 = j*64+n
  const int j = idx >> 6;
  const int n = idx & 63;
  v2f p;
  p.x = W[(size_t)(2 * j)     * N_DIM + n];
  p.y = W[(size_t)(2 * j + 1) * N_DIM + n];
  *(v2f*)(Wp + (size_t)idx * 2) = p;
}

// ---------------------------------------------------------------------------
// GEMM + bias + mask epilogue.
// ---------------------------------------------------------------------------
__global__ __launch_bounds__(256) void gemm_wmma_f32(
    const float* __restrict__ A,        // [M_TOTAL, K_DIM] row-major
    const float* __restrict__ Wp,       // [K_DIM/2, N_DIM, 2] packed pairs
    const float* __restrict__ bias,     // [N_DIM]
    const unsigned char* __restrict__ att, // [M_TOTAL] bool mask
    float* __restrict__ logits)         // [M_TOTAL, N_DIM]
{
  const int lane = threadIdx.x & 31;
  const int wave = threadIdx.x >> 5;
  const int half = lane >> 4;   // 0: lanes 0-15, 1: lanes 16-31
  const int l16  = lane & 15;
  const int m0   = blockIdx.x * ROWS_PER_BLOCK + wave * ROWS_PER_WAVE;

  // A fragment (32-bit A 16x4 layout): lane l16 = row, VGPR0/1 = {K,K+1},
  // lanes 16-31 shifted by +2 in K. One aligned b64 per lane per k-step.
  const float* Arow0 = A + (size_t)(m0 + l16) * K_DIM + 2 * half;
  const float* Arow1 = Arow0 + (size_t)16 * K_DIM;

  // B fragment base in packed W: pair index j = k/2 + half, column = l16+16t.
  // Address of pair (j, n) = Wp + (j*64 + n)*2 => base + k*64 + t*32.
  const float* Wl = Wp + ((size_t)half * N_DIM + l16) * 2;

  v8f a00 = {}, a01 = {}, a02 = {}, a03 = {};
  v8f a10 = {}, a11 = {}, a12 = {}, a13 = {};

#pragma unroll 2
  for (int k = 0; k < K_DIM; k += 4) {
    v2f af0 = *(const v2f*)(Arow0 + k);
    v2f af1 = *(const v2f*)(Arow1 + k);
    const float* wk = Wl + (size_t)k * N_DIM;      // (k/2)*64*2 floats
    v2f b0 = *(const v2f*)(wk + 0);
    v2f b1 = *(const v2f*)(wk + 32);
    v2f b2 = *(const v2f*)(wk + 64);
    v2f b3 = *(const v2f*)(wk + 96);
    // 8 args: (neg_a, A, neg_b, B, c_mod, C, reuse_a, reuse_b)
    a00 = __builtin_amdgcn_wmma_f32_16x16x4_f32(false, af0, false, b0, (short)0, a00, false, false);
    a10 = __builtin_amdgcn_wmma_f32_16x16x4_f32(false, af1, false, b0, (short)0, a10, false, false);
    a01 = __builtin_amdgcn_wmma_f32_16x16x4_f32(false, af0, false, b1, (short)0, a01, false, false);
    a11 = __builtin_amdgcn_wmma_f32_16x16x4_f32(false, af1, false, b1, (short)0, a11, false, false);
    a02 = __builtin_amdgcn_wmma_f32_16x16x4_f32(false, af0, false, b2, (short)0, a02, false, false);
    a12 = __builtin_amdgcn_wmma_f32_16x16x4_f32(false, af1, false, b2, (short)0, a12, false, false);
    a03 = __builtin_amdgcn_wmma_f32_16x16x4_f32(false, af0, false, b3, (short)0, a03, false, false);
    a13 = __builtin_amdgcn_wmma_f32_16x16x4_f32(false, af1, false, b3, (short)0, a13, false, false);
  }

  const float bv0 = bias[ 0 + l16];
  const float bv1 = bias[16 + l16];
  const float bv2 = bias[32 + l16];
  const float bv3 = bias[48 + l16];

  // C/D layout: VGPR r -> M = r + 8*half, N = l16 (+ tile*16).
#pragma unroll
  for (int r = 0; r < 8; ++r) {
    {
      const int row = m0 + 8 * half + r;
      const bool ok = att[row] != 0;
      float* orow = logits + (size_t)row * N_DIM + l16;
      orow[ 0] = ok ? a00[r] + bv0 : NEG_INF;
      orow[16] = ok ? a01[r] + bv1 : NEG_INF;
      orow[32] = ok ? a02[r] + bv2 : NEG_INF;
      orow[48] = ok ? a03[r] + bv3 : NEG_INF;
    }
    {
      const int row = m0 + 16 + 8 * half + r;
      const bool ok = att[row] != 0;
      float* orow = logits + (size_t)row * N_DIM + l16;
      orow[ 0] = ok ? a10[r] + bv0 : NEG_INF;
      orow[16] = ok ? a11[r] + bv1 : NEG_INF;
      orow[32] = ok ? a12[r] + bv2 : NEG_INF;
      orow[48] = ok ? a13[r] + bv3 : NEG_INF;
    }
  }
}

// ---------------------------------------------------------------------------
// Per-row argmax over 64 logits; -1 for masked rows (first-max tie-break).
// ---------------------------------------------------------------------------
__global__ __launch_bounds__(256) void labels_kernel(
    const float* __restrict__ logits,
    const unsigned char* __restrict__ att,
    float* __restrict__ labels)
{
  const int row = blockIdx.x * blockDim.x + threadIdx.x;
  if (row >= M_TOTAL) return;
  if (!att[row]) { labels[row] = -1.0f; return; }
  const float* lrow = logits + (size_t)row * N_DIM;
  float best = lrow[0];
  int bi = 0;
#pragma unroll
  for (int c = 1; c < N_DIM; ++c) {
    const float v = lrow[c];
    if (v > best) { best = v; bi = c; }
  }
  labels[row] = (float)bi;
}

extern "C" void kernel_launch(void* const* d_in, const int* in_sizes, int n_in,
                              void* d_out, int out_size, void* d_ws, size_t ws_size,
                              hipStream_t stream) {
  (void)in_sizes; (void)n_in; (void)out_size; (void)ws_size;
  const float*         A    = (const float*)d_in[0];         // emb [32,1024,1024]
  const unsigned char* att  = (const unsigned char*)d_in[1]; // bool [32,1024]
  const float*         W    = (const float*)d_in[2];         // [1024,64]
  const float*         bias = (const float*)d_in[3];         // [64]

  float* out    = (float*)d_out;
  float* labels = out;            // first 32768 elements (tuple output #1)
  float* logits = out + M_TOTAL;  // next 32768*64 elements (tuple output #2)
  float* Wp     = (float*)d_ws;   // 256 KB packed W

  pack_W<<<(K_DIM / 2) * N_DIM / 256, 256, 0, stream>>>(W, Wp);
  gemm_wmma_f32<<<M_TOTAL / ROWS_PER_BLOCK, 256, 0, stream>>>(A, Wp, bias, att, logits);
  labels_kernel<<<(M_TOTAL + 255) / 256, 256, 0, stream>>>(logits, att, labels);
}